// UnlearningMLP_18580028522708
// MI455X (gfx1250) — compile-verified
//
#include <hip/hip_runtime.h>

// ---------------------------------------------------------------------------
// UntrainedGCN forward for MI455X (gfx1250, wave32):
//   e = A(Ae0)  (two SpMM scatter passes, L2-resident atomics)
//   e = relu(e W0^T + b0) + e ; relu(e W1^T + b1) + e   (V_WMMA_F32_16X16X4_F32)
//   e = LayerNorm(e) * gamma + beta
// ---------------------------------------------------------------------------

typedef float v2f __attribute__((ext_vector_type(2)));
typedef float v8f __attribute__((ext_vector_type(8)));

#ifndef __has_builtin
#define __has_builtin(x) 0
#endif
#if __has_builtin(__builtin_amdgcn_wmma_f32_16x16x4_f32)
#define HAVE_WMMA_F32X4 1
#else
#define HAVE_WMMA_F32X4 0
#endif

constexpr int DIM       = 64;   // latent dim
constexpr int ROW_PAD   = 68;   // LDS row stride in floats (bank-conflict-free)
constexpr int TILE_ROWS = 128;  // rows per block (8 waves x 16 rows)
constexpr float LN_EPS  = 1e-5f;

// ---------------------------------------------------------------------------
__global__ __launch_bounds__(256) void zero_f4(float4* __restrict__ p, long long n4) {
  long long i = (long long)blockIdx.x * blockDim.x + threadIdx.x;
  if (i < n4) p[i] = make_float4(0.f, 0.f, 0.f, 0.f);
}

// One wave per edge; lane l owns columns {2l, 2l+1} of the 64-wide row.
// Gather from src[col], scatter-add into dst[row] via f32 atomics (hit in L2:
// the 51MB embedding table + edges fit the 192MB device L2).
__global__ __launch_bounds__(256) void spmm_scatter(
    const int* __restrict__ rows, const int* __restrict__ cols,
    const float* __restrict__ vals, const float* __restrict__ src,
    float* __restrict__ dst, long long E) {
  long long tid  = (long long)blockIdx.x * blockDim.x + threadIdx.x;
  long long edge = tid >> 5;
  int pair       = (int)(tid & 31);
  if (edge >= E) return;
  int   r = rows[edge];
  int   c = cols[edge];
  float v = vals[edge];
  const float2 s = *(const float2*)(src + (long long)c * DIM + pair * 2);
  float* d = dst + (long long)r * DIM + pair * 2;
  atomicAdd(d + 0, v * s.x);
  atomicAdd(d + 1, v * s.y);
}

// ---------------------------------------------------------------------------
// Fused 2-layer residual MLP + LayerNorm, in-place on e.
// Per wave: 16 rows. A fragments (16x4 f32): lane m=lane%16 holds row m,
// K pair khalf = 2*(lane/16). B fragments mirror with N=lane%16.
// C/D layout: vgpr j -> row j + 8*(lane/16), col lane%16.
__global__ __launch_bounds__(256) void mlp_ln(
    float* __restrict__ e, const float* __restrict__ Ws,
    const float* __restrict__ bs, const float* __restrict__ gammav,
    const float* __restrict__ betav, long long N) {
  __shared__ __align__(16) float sW[DIM * ROW_PAD];          // one layer of W, padded
  __shared__ __align__(16) float stage[TILE_ROWS * ROW_PAD]; // row tile, padded
  __shared__ float sBias[2 * DIM];
  __shared__ float sG[DIM];
  __shared__ float sBt[DIM];
  __shared__ float sMu[TILE_ROWS];
  __shared__ float sInv[TILE_ROWS];

  const int t     = threadIdx.x;
  const int lane  = t & 31;
  const int wave  = t >> 5;
  const int mrow  = lane & 15;          // M (or N) index inside fragment
  const int khalf = (lane >> 4) << 1;   // 0 or 2 : K-pair selector
  const int rhalf = (lane >> 4) << 3;   // 0 or 8 : C-layout row-half
  const int lrow0 = wave * 16;          // wave's first local row
  const long long base = (long long)blockIdx.x * TILE_ROWS;

  if (t < 2 * DIM) sBias[t] = bs[t];
  if (t < DIM) { sG[t] = gammav[t]; sBt[t] = betav[t]; }

  // Stage 128 rows x 64 cols into LDS (coalesced float4 loads).
#pragma unroll
  for (int i = 0; i < (TILE_ROWS * DIM) / (256 * 4); ++i) {
    int flat4 = i * 256 + t;
    int row   = flat4 >> 4;     // 16 float4 per row
    int c4    = flat4 & 15;
    long long g = base + row;
    float4 v = make_float4(0.f, 0.f, 0.f, 0.f);
    if (g < N) v = *(const float4*)(e + g * DIM + c4 * 4);
    *(float4*)(&stage[row * ROW_PAD + c4 * 4]) = v;
  }

  for (int layer = 0; layer < 2; ++layer) {
    __syncthreads();  // stage filled / previous layer done reading sW
    for (int i = t; i < DIM * DIM; i += 256) {
      int r = i >> 6, c = i & 63;
      sW[r * ROW_PAD + c] = Ws[layer * DIM * DIM + i];  // W[n][k], padded rows
    }
    __syncthreads();  // sW ready

    // A fragments: x[lrow0+m][kk*4 + khalf .. +1]  (conflict-free with pad 68)
    v2f a[16];
#pragma unroll
    for (int kk = 0; kk < 16; ++kk)
      a[kk] = *(const v2f*)&stage[(lrow0 + mrow) * ROW_PAD + kk * 4 + khalf];

    // out tile (16 x 64) = x (16 x 64) @ W^T; B[k][n] = W[n][k]
    v8f acc[4];
#pragma unroll
    for (int tc = 0; tc < 4; ++tc) {
      v8f c = {};
#pragma unroll
      for (int kk = 0; kk < 16; ++kk) {
        v2f b = *(const v2f*)&sW[(tc * 16 + mrow) * ROW_PAD + kk * 4 + khalf];
#if HAVE_WMMA_F32X4
        c = __builtin_amdgcn_wmma_f32_16x16x4_f32(false, a[kk], false, b,
                                                  (short)0, c, false, false);
#else
        c[0] += a[kk][0] * b[0];  // placeholder; expect gfx1250 WMMA path
#endif
      }
      acc[tc] = c;
    }

    // y = relu(acc + bias) + x, written back to staging in C layout.
    // Each lane touches exactly the cells it reads -> no cross-lane hazard here.
#pragma unroll
    for (int tc = 0; tc < 4; ++tc) {
      int coln   = tc * 16 + mrow;
      float bias = sBias[layer * DIM + coln];
#pragma unroll
      for (int j = 0; j < 8; ++j) {
        int row  = lrow0 + rhalf + j;
        float xv = stage[row * ROW_PAD + coln];
        stage[row * ROW_PAD + coln] = fmaxf(acc[tc][j] + bias, 0.f) + xv;
      }
    }
  }
  __syncthreads();

  // LayerNorm stats: one thread per row.
  if (t < TILE_ROWS) {
    const float* r = &stage[t * ROW_PAD];
    float s = 0.f, ss = 0.f;
#pragma unroll
    for (int c = 0; c < DIM; ++c) { float x = r[c]; s += x; ss += x * x; }
    float mu  = s * (1.f / DIM);
    float var = ss * (1.f / DIM) - mu * mu;
    sMu[t]  = mu;
    sInv[t] = rsqrtf(var + LN_EPS);
  }
  __syncthreads();

  // Normalize + affine, coalesced float4 stores.
#pragma unroll
  for (int i = 0; i < (TILE_ROWS * DIM) / (256 * 4); ++i) {
    int flat4 = i * 256 + t;
    int row   = flat4 >> 4;
    int c4    = flat4 & 15;
    long long g = base + row;
    if (g < N) {
      float mu = sMu[row], inv = sInv[row];
      const float* sp = &stage[row * ROW_PAD + c4 * 4];
      float4 v;
      v.x = (sp[0] - mu) * inv * sG[c4 * 4 + 0] + sBt[c4 * 4 + 0];
      v.y = (sp[1] - mu) * inv * sG[c4 * 4 + 1] + sBt[c4 * 4 + 1];
      v.z = (sp[2] - mu) * inv * sG[c4 * 4 + 2] + sBt[c4 * 4 + 2];
      v.w = (sp[3] - mu) * inv * sG[c4 * 4 + 3] + sBt[c4 * 4 + 3];
      *(float4*)(e + g * DIM + c4 * 4) = v;
    }
  }
}

// ---------------------------------------------------------------------------
extern "C" void kernel_launch(void* const* d_in, const int* in_sizes, int n_in,
                              void* d_out, int out_size, void* d_ws, size_t ws_size,
                              hipStream_t stream) {
  (void)n_in; (void)out_size; (void)ws_size;
  const int*   rows = (const int*)d_in[0];
  const int*   cols = (const int*)d_in[1];
  const float* vals = (const float*)d_in[2];
  const float* ini  = (const float*)d_in[3];
  const float* Ws   = (const float*)d_in[4];
  const float* bs   = (const float*)d_in[5];
  const float* gam  = (const float*)d_in[6];
  const float* bet  = (const float*)d_in[7];
  float* out  = (float*)d_out;
  float* bufA = (float*)d_ws;   // N*64 floats of scratch for hop-1 result

  const long long E     = in_sizes[0];
  const long long N     = (long long)in_sizes[3] / DIM;
  const long long total = N * DIM;

  const long long n4 = total / 4;                      // DIM multiple of 4
  const int zb = (int)((n4 + 255) / 256);
  const int sb = (int)((E * 32 + 255) / 256);          // one wave per edge
  const int mb = (int)((N + TILE_ROWS - 1) / TILE_ROWS);

  zero_f4     <<<zb, 256, 0, stream>>>((float4*)bufA, n4);
  spmm_scatter<<<sb, 256, 0, stream>>>(rows, cols, vals, ini,  bufA, E);
  zero_f4     <<<zb, 256, 0, stream>>>((float4*)out, n4);
  spmm_scatter<<<sb, 256, 0, stream>>>(rows, cols, vals, bufA, out,  E);
  mlp_ln      <<<mb, 256, 0, stream>>>(out, Ws, bs, gam, bet, N);
}